// GADNRBase_90426241450737
// MI455X (gfx1250) — compile-verified
//
#include <hip/hip_runtime.h>
#include <cstdint>
#include <cstddef>

// ---------------------------------------------------------------------------
// GAD-NR forward for MI455X (gfx1250, wave32).
// Memory-bound workload (~25 GFLOP f32, hot data L2-resident) -> keep f32,
// use V_WMMA_F32_16X16X4_F32 for all dense layers, TDM tensor_load_to_lds
// (with pad_enable for conflict-free LDS strides) for operand staging.
// Rank-1 structure of target/gen covariances lets det/inv/trace collapse to
// 32-wide dot products (matrix determinant lemma + Sherman-Morrison).
// ---------------------------------------------------------------------------

#define NN     50000
#define NE     1600000
#define FDIM   128
#define HID    32
#define NPAD   50048      // 128-row padded
#define R2     100000     // S*N generator rows
#define R2PAD  100096

typedef __attribute__((ext_vector_type(2))) float    v2f;
typedef __attribute__((ext_vector_type(8))) float    v8f;
typedef __attribute__((ext_vector_type(4))) unsigned v4u;
typedef __attribute__((ext_vector_type(4))) int      v4i;
typedef __attribute__((ext_vector_type(8))) int      v8i;

// ---------------------------------------------------------------------------
// Tensor Data Mover: 2D tile (width x tile_h elems of f32) from global to LDS.
// tensor_h < tile_h rows are zero-filled by TDM OOB handling (ragged tails).
// pad_enable inserts 1 DWORD after every `width` DWORDs -> LDS stride width+1.
// ---------------------------------------------------------------------------
__device__ __forceinline__ void tdm_load_2d(unsigned lds_byte_off, const void* gptr,
                                            unsigned width, unsigned tile_h,
                                            unsigned tensor_h, unsigned stride_elems,
                                            unsigned pad_interval_code) {
  unsigned long long ga = (unsigned long long)(uintptr_t)gptr;
  v4u g0 = (v4u)0u;
  g0[0] = 1u;                                             // count=1, user mode
  g0[1] = lds_byte_off;                                   // lds_addr
  g0[2] = (unsigned)(ga & 0xFFFFFFFFull);                 // global_addr lo
  g0[3] = (unsigned)((ga >> 32) & 0x1FFFFFFull) | (2u << 30);  // addr hi | type=2
  v8i g1 = (v8i)0;
  unsigned d0 = (2u << 16)                                // data_size = 4B
              | (1u << 20)                                // pad_enable
              | (pad_interval_code << 22);                // pad every 2^(c+1) DW
                                                          // pad_amount code 0 = 1 DW
  g1[0] = (int)d0;
  g1[1] = (int)((width & 0xFFFFu) << 16);                               // tdim0 lo16
  g1[2] = (int)(((width >> 16) & 0xFFFFu) | ((tensor_h & 0xFFFFu) << 16)); // tdim0 hi | tdim1 lo
  g1[3] = (int)(((tensor_h >> 16) & 0xFFFFu) | ((width & 0xFFFFu) << 16)); // tdim1 hi | tile_dim0
  g1[4] = (int)(tile_h & 0xFFFFu);                                       // tile_dim1 (tile_dim2=0)
  g1[5] = (int)stride_elems;                                             // dim0_stride lo32
  g1[6] = 0;
  g1[7] = 0;
  v4i gz = (v4i)0;
#if __clang_major__ >= 23
  v8i gz8 = (v8i)0;
  __builtin_amdgcn_tensor_load_to_lds(g0, g1, gz, gz, gz8, 0);
#else
  __builtin_amdgcn_tensor_load_to_lds(g0, g1, gz, gz, 0);
#endif
}

// ---------------------------------------------------------------------------
// Generic WMMA f32 node GEMM:  out[n, 0..CT*16) = act(A[n,0..K) @ W^T + bias)
// W row-major [CT*16, K] (PyTorch Linear layout). One 16-row tile per wave.
// A and W staged to LDS by TDM (wave 0) with stride K+1.
// ACT: 0=none 1=relu 2=tanh.  ACC: out += result.
// ---------------------------------------------------------------------------
template<int K, int CT, int ACT, bool ACC>
__global__ __launch_bounds__(256) void k_gemm(const float* __restrict__ A,
                                              const float* __restrict__ W,
                                              const float* __restrict__ bias,
                                              float* __restrict__ out, int nrows) {
  extern __shared__ float smem[];
  constexpr int LDP = K + 1;
  constexpr unsigned ICODE = (K == 32) ? 4u : (K == 64) ? 5u : 6u;
  float* Ws = smem;                 // [CT*16][LDP]
  float* As = smem + CT * 16 * LDP; // [128][LDP]
  const int block0 = blockIdx.x * 128;
  if (threadIdx.x < 32) {
    tdm_load_2d(0u, W, K, CT * 16, CT * 16, K, ICODE);
    int remain = nrows - block0;
    tdm_load_2d((unsigned)(CT * 16 * LDP * 4), A + (size_t)block0 * K,
                K, 128, (unsigned)(remain > 0 ? remain : 0), K, ICODE);
    __builtin_amdgcn_s_wait_tensorcnt(0);
  }
  __syncthreads();

  const int wv = threadIdx.x >> 5;
  const int l  = threadIdx.x & 31;
  const int half = l >> 4, lane16 = l & 15;
  const int row0 = wv * 16;

  v8f acc[CT];
#pragma unroll
  for (int c = 0; c < CT; ++c) acc[c] = (v8f)0.0f;

  for (int k = 0; k < K; k += 4) {
    const int r0 = k + half * 2;
    v2f a;
    a.x = As[(row0 + lane16) * LDP + r0];
    a.y = As[(row0 + lane16) * LDP + r0 + 1];
#pragma unroll
    for (int c = 0; c < CT; ++c) {
      v2f b;
      b.x = Ws[(c * 16 + lane16) * LDP + r0];
      b.y = Ws[(c * 16 + lane16) * LDP + r0 + 1];
      acc[c] = __builtin_amdgcn_wmma_f32_16x16x4_f32(false, a, false, b,
                                                     (short)0, acc[c], false, false);
    }
  }

  const int grb = block0 + row0 + half * 8;
#pragma unroll
  for (int v = 0; v < 8; ++v) {
    const int gr = grb + v;
    if (gr >= nrows) continue;
#pragma unroll
    for (int c = 0; c < CT; ++c) {
      const int col = c * 16 + lane16;
      float x = acc[c][v];
      if (bias) x += bias[col];
      if (ACT == 1) x = fmaxf(x, 0.0f);
      else if (ACT == 2) x = tanhf(x);
      float* p = out + (size_t)gr * (CT * 16) + col;
      if (ACC) *p += x; else *p = x;
    }
  }
}

// ---------------------------------------------------------------------------
// Edge segment sum: seg[dst] += h[src] (one lane per feature), cnt[dst] += 1.
// ---------------------------------------------------------------------------
__global__ __launch_bounds__(256) void k_seg(const float* __restrict__ h,
                                             const int* __restrict__ src,
                                             const int* __restrict__ dst,
                                             float* __restrict__ seg,
                                             float* __restrict__ cnt) {
  const long tid = (long)blockIdx.x * 256 + threadIdx.x;
  const long e = tid >> 5;
  if (e >= NE) return;
  const int l = (int)(tid & 31);
  if (l == 0) __builtin_prefetch(&src[e + 2048], 0, 1);
  const int s = src[e], d = dst[e];
  atomicAdd(&seg[(size_t)d * HID + l], h[(size_t)s * HID + l]);
  if (cnt && l == 0) atomicAdd(&cnt[d], 1.0f);
}

// ---------------------------------------------------------------------------
// PNA edge GEMM: he = [h0[dst]||h0[src]] @ Wpre^T + bpre (16 edges per wave,
// WMMA K=64), then scatter-add he / he^2 into m1/m2 at dst.
// ---------------------------------------------------------------------------
__global__ __launch_bounds__(256) void k_pna_edge(const float* __restrict__ h0,
                                                  const int* __restrict__ src,
                                                  const int* __restrict__ dst,
                                                  const float* __restrict__ W,
                                                  const float* __restrict__ bias,
                                                  float* __restrict__ m1,
                                                  float* __restrict__ m2) {
  extern __shared__ float smem[];
  constexpr int LDP = 65;
  float* Ws = smem;              // [32][65]
  float* As = smem + 32 * LDP;   // [128][65]
  const int e0b = blockIdx.x * 128;
  if (threadIdx.x < 32) {
    tdm_load_2d(0u, W, 64, 32, 32, 64, 5u);
    __builtin_amdgcn_s_wait_tensorcnt(0);
  }
  const int wv = threadIdx.x >> 5;
  const int l  = threadIdx.x & 31;
  for (int e = wv; e < 128; e += 8) {
    const int ss = src[e0b + e], dd = dst[e0b + e];
    As[e * LDP + l]      = h0[(size_t)dd * HID + l];
    As[e * LDP + 32 + l] = h0[(size_t)ss * HID + l];
  }
  __syncthreads();

  const int half = l >> 4, lane16 = l & 15;
  const int row0 = wv * 16;
  v8f acc[2];
  acc[0] = (v8f)0.0f; acc[1] = (v8f)0.0f;
  for (int k = 0; k < 64; k += 4) {
    const int r0 = k + half * 2;
    v2f a;
    a.x = As[(row0 + lane16) * LDP + r0];
    a.y = As[(row0 + lane16) * LDP + r0 + 1];
#pragma unroll
    for (int c = 0; c < 2; ++c) {
      v2f b;
      b.x = Ws[(c * 16 + lane16) * LDP + r0];
      b.y = Ws[(c * 16 + lane16) * LDP + r0 + 1];
      acc[c] = __builtin_amdgcn_wmma_f32_16x16x4_f32(false, a, false, b,
                                                     (short)0, acc[c], false, false);
    }
  }
  const int erb = e0b + row0 + half * 8;
#pragma unroll
  for (int v = 0; v < 8; ++v) {
    const int e = erb + v;
    const int d = dst[e];
#pragma unroll
    for (int c = 0; c < 2; ++c) {
      const int col = c * 16 + lane16;
      const float he = acc[c][v] + bias[col];
      atomicAdd(&m1[(size_t)d * HID + col], he);
      atomicAdd(&m2[(size_t)d * HID + col], he * he);
    }
  }
}

// --------------------------- elementwise kernels ---------------------------
__global__ void k_add(const float* a, const float* b, float* o, int n) {
  int i = blockIdx.x * 256 + threadIdx.x;
  if (i < n) o[i] = a[i] + b[i];
}
__global__ void k_meannb(const float* seg, const float* cnt, float* o, int n) {
  int i = blockIdx.x * 256 + threadIdx.x;
  if (i < n) o[i] = seg[i] / fmaxf(cnt[i >> 5], 1.0f);
}
__global__ void k_stdagg(const float* m1, const float* m2, const float* cnt,
                         float* o, int n) {
  int i = blockIdx.x * 256 + threadIdx.x;
  if (i >= n) return;
  float c = fmaxf(cnt[i >> 5], 1.0f);
  float a = m1[i] / c, b = m2[i] / c;
  o[i] = sqrtf(fmaxf(b - a * a, 0.0f) + 1e-5f);
}
__global__ void k_concat(const float* h0, const float* sa, float* o, int n) {
  int i = blockIdx.x * 256 + threadIdx.x;
  if (i >= n) return;
  int node = i >> 6, j = i & 63;
  o[i] = (j < 32) ? h0[node * HID + j] : sa[node * HID + (j - 32)];
}
__global__ void k_bcast3(const float* fr, float* o, int n) {
  int i = blockIdx.x * 256 + threadIdx.x;
  if (i >= n) return;
  float v = fr[i];
  o[i] = v; o[n + i] = v; o[2 * n + i] = v;
}
__global__ void k_target(const float* stdn, float* dett) {
  int nid = blockIdx.x * 256 + threadIdx.x;
  if (nid >= NN) return;
  float s2 = 0.f;
  for (int h = 0; h < HID; ++h) { float s = stdn[nid * HID + h]; s2 += s * s; }
  float d = 1.0f + s2;  // det(sigma sigma^T + I), matrix determinant lemma
  dett[nid] = d; dett[NN + nid] = d; dett[2 * NN + nid] = d;
}
__global__ void k_deg_head(const float* d3, const float* W3, const float* b3,
                           const float* w1, const float* b1, float* o) {
  int nid = blockIdx.x * 256 + threadIdx.x;
  if (nid >= NN) return;
  float acc = b3[0];
  for (int h = 0; h < HID; ++h) acc += d3[nid * HID + h] * W3[h];
  float r = fmaxf(acc, 0.0f);
  o[nid] = fmaxf(r * w1[0] + b1[0], 0.0f);
}

// deterministic hash Box-Muller stand-in for jax.random.normal
__device__ __forceinline__ unsigned hash_u32(unsigned x) {
  x ^= x >> 16; x *= 0x7FEB352Du; x ^= x >> 15; x *= 0x846CA68Bu; x ^= x >> 16;
  return x;
}
__device__ __forceinline__ float hash_normal(unsigned idx) {
  unsigned h1 = hash_u32(idx * 2u + 1u);
  unsigned h2 = hash_u32(idx * 2654435761u + 0x9E3779B9u);
  float u1 = ((h1 >> 8) + 0.5f) * (1.0f / 16777216.0f);
  float u2 = (h2 >> 8) * (1.0f / 16777216.0f);
  return sqrtf(-2.0f * logf(u1)) * cosf(6.28318530718f * u2);
}
__global__ void k_prep_var(const float* gmean, const float* gsig, float* var, int t) {
  int i = blockIdx.x * 256 + threadIdx.x;        // over 2*NN*HID
  if (i >= 2 * NN * HID) return;
  int s = i / (NN * HID);
  int rem = i - s * (NN * HID);
  unsigned seed = (unsigned)(((t * 2 + s) * (NN * HID)) + rem);
  var[i] = gmean[rem] + expf(gsig[rem]) * hash_normal(seed);
}

// Per-node covariance terms: Sherman-Morrison closed form.
// gen_cov = u u^T + I with u = |a-b|/2 ;  T = sigma sigma^T + I.
__global__ void k_stats(const float* nh, const float* stdn, const float* meann,
                        float* detg, float* tr, float* zq) {
  int nid = blockIdx.x * 256 + threadIdx.x;
  if (nid >= NN) return;
  float su2 = 0.f, ss2 = 0.f, us = 0.f, ud = 0.f, dd = 0.f;
  for (int h = 0; h < HID; ++h) {
    float a = nh[(size_t)nid * HID + h];
    float b = nh[(size_t)(NN + nid) * HID + h];
    float gm = 0.5f * (a + b);
    float u  = 0.5f * fabsf(a - b);   // gs/sqrt(S), S=2
    float sg = stdn[nid * HID + h];
    float d  = gm - meann[nid * HID + h];
    su2 += u * u; ss2 += sg * sg; us += u * sg; ud += u * d; dd += d * d;
  }
  float alpha = 1.0f / (1.0f + su2);
  detg[nid] = 1.0f + su2;
  tr[nid]   = ss2 + (float)HID - alpha * (us * us + su2);
  zq[nid]   = dd - alpha * ud * ud;
}

// ---------------------------------------------------------------------------
#define GEMM(Kv, CTv, ACTv, ACCv, Ap, Wp, Bp, Op, NR)                          \
  do {                                                                         \
    int _nb = ((NR) + 127) / 128;                                              \
    size_t _sh = ((size_t)((CTv)*16) * ((Kv) + 1) + 128u * ((Kv) + 1)) * 4u;   \
    k_gemm<Kv, CTv, ACTv, ACCv><<<_nb, 256, _sh, stream>>>((Ap), (Wp), (Bp),   \
                                                           (Op), (NR));        \
  } while (0)

extern "C" void kernel_launch(void* const* d_in, const int* in_sizes, int n_in,
                              void* d_out, int out_size, void* d_ws, size_t ws_size,
                              hipStream_t stream) {
  (void)in_sizes; (void)n_in; (void)out_size; (void)ws_size;
  auto P = [&](int i) { return (const float*)d_in[i]; };
  const float* x   = P(0);
  const int*   ei  = (const int*)d_in[1];
  const int*   src = ei;
  const int*   dst = ei + NE;
  float* o = (float*)d_out;

  // output layout (floats)
  const size_t OFF_H0 = 0, OFF_L1 = (size_t)NN * HID, OFF_DEG = 2ull * NN * HID;
  const size_t OFF_FEA = OFF_DEG + NN;
  const size_t OFF_DETT = OFF_FEA + 3ull * NN * FDIM;
  const size_t OFF_DETG = OFF_DETT + 3ull * NN;
  const size_t OFF_TR   = OFF_DETG + 3ull * NN;
  const size_t OFF_ZQ   = OFF_TR + 3ull * NN;

  // workspace
  float* w = (float*)d_ws;
  size_t off = 0;
  auto alloc = [&](size_t n) { float* p = w + off; off += n; return p; };
  float* bh0   = alloc((size_t)NPAD * HID);
  float* seg0  = alloc((size_t)NPAD * HID);
  float* seg1  = alloc((size_t)NPAD * HID);
  float* cnt   = alloc(NPAD);
  float* tmpA  = alloc((size_t)NPAD * HID);
  float* tmpB  = alloc((size_t)NPAD * HID);
  float* br1   = alloc((size_t)NPAD * HID);
  float* bl1   = alloc((size_t)NPAD * HID);
  float* meanN = alloc((size_t)NPAD * HID);
  float* m1b   = alloc((size_t)NPAD * HID);
  float* m2b   = alloc((size_t)NPAD * HID);
  float* sagg  = alloc((size_t)NPAD * HID);
  float* cat   = alloc((size_t)NPAD * 64);
  float* stdN  = alloc((size_t)NPAD * HID);
  float* gmean = alloc((size_t)NPAD * HID);
  float* gsig  = alloc((size_t)NPAD * HID);
  float* f3    = alloc((size_t)NPAD * FDIM);
  float* fr    = alloc((size_t)NPAD * FDIM);
  float* var   = alloc((size_t)R2PAD * HID);
  float* nhA   = alloc((size_t)R2PAD * HID);
  float* nhB   = alloc((size_t)R2PAD * HID);

  hipMemsetAsync(seg0, 0, (size_t)NPAD * HID * 4, stream);
  hipMemsetAsync(seg1, 0, (size_t)NPAD * HID * 4, stream);
  hipMemsetAsync(cnt, 0, (size_t)NPAD * 4, stream);
  hipMemsetAsync(m1b, 0, (size_t)NPAD * HID * 4, stream);
  hipMemsetAsync(m2b, 0, (size_t)NPAD * HID * 4, stream);

  const int NEl = NN * HID;
  const int ebN = (NEl + 255) / 256;
  const int segBlocks = (int)(((long)NE * 32 + 255) / 256);

  // h0 = x @ Wl^T + bl
  GEMM(128, 2, 0, false, x, P(2), P(3), bh0, NN);
  hipMemcpyAsync(o + OFF_H0, bh0, (size_t)NEl * 4, hipMemcpyDeviceToDevice, stream);

  // GIN layer 1:  r1 = relu( relu((h0+seg0)@W1+b1) @ W2 + b2 )
  k_seg<<<segBlocks, 256, 0, stream>>>(bh0, src, dst, seg0, cnt);
  k_add<<<ebN, 256, 0, stream>>>(bh0, seg0, tmpA, NEl);
  GEMM(32, 2, 1, false, tmpA, P(4), P(5), tmpB, NN);
  GEMM(32, 2, 1, false, tmpB, P(6), P(7), br1, NN);

  // GIN layer 2 -> l1
  k_seg<<<segBlocks, 256, 0, stream>>>(br1, src, dst, seg1, (float*)nullptr);
  k_add<<<ebN, 256, 0, stream>>>(br1, seg1, tmpA, NEl);
  GEMM(32, 2, 1, false, tmpA, P(8), P(9), tmpB, NN);
  GEMM(32, 2, 0, false, tmpB, P(10), P(11), bl1, NN);
  hipMemcpyAsync(o + OFF_L1, bl1, (size_t)NEl * 4, hipMemcpyDeviceToDevice, stream);

  // degree decoder
  GEMM(32, 2, 2, false, bl1, P(12), P(13), tmpA, NN);
  GEMM(32, 2, 2, false, tmpA, P(14), P(15), tmpB, NN);
  GEMM(32, 2, 2, false, tmpB, P(16), P(17), tmpA, NN);
  k_deg_head<<<(NN + 255) / 256, 256, 0, stream>>>(tmpA, P(18), P(19), P(20), P(21),
                                                   o + OFF_DEG);

  // mean_neigh = SAGE(mean_nb) + h0 @ sage_r^T
  k_meannb<<<ebN, 256, 0, stream>>>(seg0, cnt, tmpB, NEl);
  GEMM(32, 2, 0, false, tmpB, P(54), P(55), meanN, NN);
  GEMM(32, 2, 0, true, bh0, P(56), (const float*)nullptr, meanN, NN);

  // PNA std aggregation -> std_neigh
  {
    int nb = NE / 128;
    size_t sh = (size_t)(32 * 65 + 128 * 65) * 4;
    k_pna_edge<<<nb, 256, sh, stream>>>(bh0, src, dst, P(57), P(58), m1b, m2b);
  }
  k_stdagg<<<ebN, 256, 0, stream>>>(m1b, m2b, cnt, sagg, NEl);
  k_concat<<<(NN * 64 + 255) / 256, 256, 0, stream>>>(bh0, sagg, cat, NN * 64);
  GEMM(64, 2, 0, false, cat, P(59), P(60), tmpA, NN);
  GEMM(32, 2, 0, false, tmpA, P(61), P(62), stdN, NN);
  k_target<<<(NN + 255) / 256, 256, 0, stream>>>(stdN, o + OFF_DETT);

  // gen_mean_base / gen_sig_base
  GEMM(32, 2, 2, false, bl1, P(30), P(31), tmpA, NN);
  GEMM(32, 2, 2, false, tmpA, P(32), P(33), tmpB, NN);
  GEMM(32, 2, 1, false, tmpB, P(34), P(35), tmpA, NN);
  GEMM(32, 2, 0, false, tmpA, P(36), P(37), gmean, NN);
  GEMM(32, 2, 2, false, bl1, P(38), P(39), tmpA, NN);
  GEMM(32, 2, 2, false, tmpA, P(40), P(41), tmpB, NN);
  GEMM(32, 2, 1, false, tmpB, P(42), P(43), tmpA, NN);
  GEMM(32, 2, 0, false, tmpA, P(44), P(45), gsig, NN);

  // feature decoder: tanh,tanh, (32->128)+relu, 128x128 linear, broadcast x3
  GEMM(32, 2, 2, false, bl1, P(22), P(23), tmpA, NN);
  GEMM(32, 2, 2, false, tmpA, P(24), P(25), tmpB, NN);
  GEMM(32, 8, 1, false, tmpB, P(26), P(27), f3, NN);
  GEMM(128, 8, 0, false, f3, P(28), P(29), fr, NN);
  k_bcast3<<<(NN * FDIM + 255) / 256, 256, 0, stream>>>(fr, o + OFF_FEA, NN * FDIM);

  // generator sampling loop
  for (int t = 0; t < 3; ++t) {
    k_prep_var<<<(2 * NEl + 255) / 256, 256, 0, stream>>>(gmean, gsig, var, t);
    GEMM(32, 2, 1, false, var, P(46), P(47), nhA, R2);
    GEMM(32, 2, 1, false, nhA, P(48), P(49), nhB, R2);
    GEMM(32, 2, 1, false, nhB, P(50), P(51), nhA, R2);
    GEMM(32, 2, 0, false, nhA, P(52), P(53), nhB, R2);
    k_stats<<<(NN + 255) / 256, 256, 0, stream>>>(nhB, stdN, meanN,
                                                  o + OFF_DETG + (size_t)t * NN,
                                                  o + OFF_TR + (size_t)t * NN,
                                                  o + OFF_ZQ + (size_t)t * NN);
  }
}